// LaplacianPE_38654705664484
// MI455X (gfx1250) — compile-verified
//
#include <hip/hip_runtime.h>
#include <hip/hip_bf16.h>
#include <math.h>

// ---------------- problem constants ----------------
#define NN   4096      // num nodes
#define NE   131072    // num edges
#define BS   16        // subspace block size
#define KPE  8         // positional-encoding dim
#define HID  64        // hidden dim
#define ITERS 120      // subspace-iteration count

typedef __attribute__((ext_vector_type(2))) float v2f;
typedef __attribute__((ext_vector_type(8))) float v8f;

// ---------------- small utility kernels ----------------
__global__ void lappe_zerof(float* p, int n) {
  int i = blockIdx.x * blockDim.x + threadIdx.x;
  if (i < n) p[i] = 0.0f;
}

__global__ void lappe_copyf(const float* __restrict__ src, float* __restrict__ dst, int n) {
  int i = blockIdx.x * blockDim.x + threadIdx.x;
  if (i < n) dst[i] = src[i];
}

// deg[col[e]] += 1
__global__ void lappe_deg(const int* __restrict__ ei, float* deg) {
  int e = blockIdx.x * blockDim.x + threadIdx.x;
  if (e < NE) atomicAdd(&deg[ei[NE + e]], 1.0f);
}

// dis = clip(deg,1)^-0.5
__global__ void lappe_dis(const float* __restrict__ deg, float* dis) {
  int i = blockIdx.x * blockDim.x + threadIdx.x;
  if (i < NN) dis[i] = rsqrtf(fmaxf(deg[i], 1.0f));
}

// deterministic pseudo-random init of the N x 16 subspace block
__global__ void lappe_initq(float* Q) {
  int i = blockIdx.x * blockDim.x + threadIdx.x;  // 0 .. NN*BS-1
  unsigned x = (unsigned)i * 2654435761u + 0x9E3779B9u;
  x ^= x >> 16; x *= 0x85EBCA6Bu; x ^= x >> 13; x *= 0xC2B2AE35u; x ^= x >> 16;
  Q[i] = (float)(x & 0xFFFFFF) * (1.0f / 16777216.0f) - 0.5f;
}

// Z += 0.5*dis[r]*dis[c] * (Q[c,:] scattered to r and Q[r,:] scattered to c)
// (symmetrized M = 2I - L; the "I" part comes from Z being pre-copied from Q)
__global__ void lappe_spmv(const int* __restrict__ ei, const float* __restrict__ dis,
                           const float* __restrict__ Q, float* Z) {
  int t = blockIdx.x * blockDim.x + threadIdx.x;  // NE*BS threads
  int e = t >> 4;
  int j = t & 15;
  int r = ei[e];
  int c = ei[NE + e];
  float w = 0.5f * dis[r] * dis[c];
  atomicAdd(&Z[r * BS + j], w * Q[c * BS + j]);
  atomicAdd(&Z[c * BS + j], w * Q[r * BS + j]);
}

// G[a,b] += sum_i A[i,a]*B[i,b]   (16x16 Gram / Rayleigh numerator)
// Tiles staged global -> LDS with CDNA5 async-to-LDS DMA (ASYNCcnt-tracked).
__global__ void lappe_gram2(const float* __restrict__ A, const float* __restrict__ B, float* G) {
  __shared__ float ta[128 * BS];
  __shared__ float tb[128 * BS];
  int base = blockIdx.x * 128 * BS;
  const float* Ab = A + base;
  const float* Bb = B + base;
  // 2048 floats per tile = 512 x b128; 256 threads -> 2 iterations, no divergence.
  for (int t = threadIdx.x; t < 512; t += 256) {
    unsigned la = (unsigned)(size_t)(&ta[4 * t]);   // LDS offset = low 32b of flat addr
    unsigned lb = (unsigned)(size_t)(&tb[4 * t]);
    unsigned long long ga = (unsigned long long)(size_t)(Ab + 4 * t);
    unsigned long long gb = (unsigned long long)(size_t)(Bb + 4 * t);
    asm volatile("global_load_async_to_lds_b128 %0, %1, off"
                 :: "v"(la), "v"(ga) : "memory");
    asm volatile("global_load_async_to_lds_b128 %0, %1, off"
                 :: "v"(lb), "v"(gb) : "memory");
  }
  asm volatile("s_wait_asynccnt 0x0" ::: "memory");  // my DMAs landed in LDS
  __syncthreads();                                   // everyone's DMAs visible
  int a = threadIdx.x >> 4;
  int b = threadIdx.x & 15;
  float acc = 0.0f;
  for (int r = 0; r < 128; ++r) acc += ta[r * BS + a] * tb[r * BS + b];
  atomicAdd(&G[a * BS + b], acc);
}

// Cholesky of 16x16 G, then store Uinv with Uinv[k][j] = (L^-1)[j][k] (upper)
__global__ void lappe_cholinv(const float* __restrict__ G, float* Uinv) {
  float A[BS][BS], L[BS][BS], Li[BS][BS];
  for (int i = 0; i < BS; ++i)
    for (int j = 0; j < BS; ++j) { A[i][j] = G[i * BS + j]; L[i][j] = 0.0f; Li[i][j] = 0.0f; }
  for (int k = 0; k < BS; ++k) {
    float d = A[k][k];
    for (int m = 0; m < k; ++m) d -= L[k][m] * L[k][m];
    d = fmaxf(d, 1e-24f);
    L[k][k] = sqrtf(d);
    float inv = 1.0f / L[k][k];
    for (int i = k + 1; i < BS; ++i) {
      float s = A[i][k];
      for (int m = 0; m < k; ++m) s -= L[i][m] * L[k][m];
      L[i][k] = s * inv;
    }
  }
  for (int j = 0; j < BS; ++j) {
    Li[j][j] = 1.0f / L[j][j];
    for (int i = j + 1; i < BS; ++i) {
      float s = 0.0f;
      for (int m = j; m < i; ++m) s += L[i][m] * Li[m][j];
      Li[i][j] = -s / L[i][i];
    }
  }
  for (int k = 0; k < BS; ++k)
    for (int j = 0; j < BS; ++j)
      Uinv[k * BS + j] = (k <= j) ? Li[j][k] : 0.0f;
}

// Q[i,:] = Z[i,:] * Uinv   (triangular apply; orthonormalizes the block)
__global__ void lappe_apply(const float* __restrict__ Z, const float* __restrict__ Uinv, float* Q) {
  __shared__ float U[BS * BS];
  if (threadIdx.x < BS * BS) U[threadIdx.x] = Uinv[threadIdx.x];
  __syncthreads();
  int i = blockIdx.x * blockDim.x + threadIdx.x;  // NN threads
  float z[BS];
  for (int k = 0; k < BS; ++k) z[k] = Z[i * BS + k];
  for (int j = 0; j < BS; ++j) {
    float acc = 0.0f;
    for (int k = 0; k <= j; ++k) acc += z[k] * U[k * BS + j];
    Q[i * BS + j] = acc;
  }
}

// Rayleigh-Ritz: T = 2I - sym(Tp); Jacobi eigensolve; take eigvecs 1..8 (ascending)
__global__ void lappe_rrsolve(const float* __restrict__ Tp, float* Ysel) {
  float A[BS][BS], V[BS][BS];
  for (int i = 0; i < BS; ++i)
    for (int j = 0; j < BS; ++j) {
      A[i][j] = ((i == j) ? 2.0f : 0.0f) - 0.5f * (Tp[i * BS + j] + Tp[j * BS + i]);
      V[i][j] = (i == j) ? 1.0f : 0.0f;
    }
  for (int sweep = 0; sweep < 30; ++sweep) {
    for (int p = 0; p < BS - 1; ++p) {
      for (int q = p + 1; q < BS; ++q) {
        float apq = A[p][q];
        if (fabsf(apq) < 1e-13f) continue;
        float theta = 0.5f * (A[q][q] - A[p][p]) / apq;
        float t = ((theta >= 0.0f) ? 1.0f : -1.0f) / (fabsf(theta) + sqrtf(1.0f + theta * theta));
        float c = rsqrtf(1.0f + t * t);
        float s = t * c;
        for (int i = 0; i < BS; ++i) { float x = A[i][p], y = A[i][q]; A[i][p] = c * x - s * y; A[i][q] = s * x + c * y; }
        for (int i = 0; i < BS; ++i) { float x = A[p][i], y = A[q][i]; A[p][i] = c * x - s * y; A[q][i] = s * x + c * y; }
        for (int i = 0; i < BS; ++i) { float x = V[i][p], y = V[i][q]; V[i][p] = c * x - s * y; V[i][q] = s * x + c * y; }
      }
    }
  }
  float ev[BS]; int idx[BS];
  for (int i = 0; i < BS; ++i) { ev[i] = A[i][i]; idx[i] = i; }
  for (int i = 1; i < BS; ++i) {            // insertion sort ascending
    float e = ev[i]; int id = idx[i]; int j = i - 1;
    while (j >= 0 && ev[j] > e) { ev[j + 1] = ev[j]; idx[j + 1] = idx[j]; --j; }
    ev[j + 1] = e; idx[j + 1] = id;
  }
  for (int j = 0; j < KPE; ++j) {
    int cs = idx[j + 1];                    // skip the null-space vector
    float mx = 0.0f; int am = 0;
    for (int i = 0; i < BS; ++i) if (fabsf(V[i][cs]) > mx) { mx = fabsf(V[i][cs]); am = i; }
    float sgn = (V[am][cs] >= 0.0f) ? 1.0f : -1.0f;   // deterministic sign convention
    for (int i = 0; i < BS; ++i) Ysel[i * KPE + j] = sgn * V[i][cs];
  }
}

// pe = Q @ Ysel; pe2 = pe * tanh(pe @ W_sign + b_sign)
__global__ void lappe_signnet(const float* __restrict__ Q, const float* __restrict__ Ysel,
                              const float* __restrict__ Wsign, const float* __restrict__ bsign,
                              float* pe2) {
  __shared__ float Ys[BS * KPE];
  __shared__ float Ws[KPE * KPE];
  __shared__ float bs[KPE];
  if (threadIdx.x < BS * KPE) Ys[threadIdx.x] = Ysel[threadIdx.x];
  if (threadIdx.x < KPE * KPE) Ws[threadIdx.x] = Wsign[threadIdx.x];
  if (threadIdx.x < KPE) bs[threadIdx.x] = bsign[threadIdx.x];
  __syncthreads();
  int i = blockIdx.x * blockDim.x + threadIdx.x;  // NN threads
  float q[BS];
  for (int k = 0; k < BS; ++k) q[k] = Q[i * BS + k];
  float p[KPE];
  for (int j = 0; j < KPE; ++j) {
    float acc = 0.0f;
    for (int k = 0; k < BS; ++k) acc += q[k] * Ys[k * KPE + j];
    p[j] = acc;
  }
  for (int j = 0; j < KPE; ++j) {
    float s = bs[j];
    for (int k = 0; k < KPE; ++k) s += p[k] * Ws[k * KPE + j];
    pe2[i * KPE + j] = p[j] * tanhf(s);
  }
}

// ---------------- WMMA fp32 GEMM + bias (+ optional exact gelu) ----------------
// D[M x Ncols] = act(A[M x Kdim] @ B[Kdim x Ncols] + bias).  One wave32 per 16x16 tile,
// V_WMMA_F32_16X16X4_F32 with a K-loop of Kdim/4 steps.
__global__ void lappe_wmma_gemm(const float* __restrict__ A, const float* __restrict__ B,
                                const float* __restrict__ bias, float* __restrict__ D,
                                int Kdim, int Ncols, int do_gelu) {
  int wave = threadIdx.x >> 5;
  int lane = threadIdx.x & 31;
  int tile = blockIdx.x * (blockDim.x >> 5) + wave;
  int nt = Ncols >> 4;
  int tm = tile / nt;
  int tn = tile - tm * nt;
  int lrow  = lane & 15;
  int half2 = (lane >> 4) << 1;            // 0 for lanes 0-15, 2 for lanes 16-31
  int row   = tm * 16 + lrow;              // A-fragment row (M = lane % 16)
  int coln  = tn * 16 + lrow;              // B/C/D column (N = lane % 16)

  __builtin_prefetch(&A[row * Kdim], 0, 3);   // global_prefetch_b8

  v8f acc = {};
  for (int k0 = 0; k0 < Kdim; k0 += 4) {
    v2f a, b;
    // A 16x4 fp32 fragment: lane holds (row, k0+half2) and (row, k0+half2+1)
    a.x = A[row * Kdim + k0 + half2];
    a.y = A[row * Kdim + k0 + half2 + 1];
    // B 4x16 fp32 fragment: lane holds (k0+half2, col) and (k0+half2+1, col)
    b.x = B[(k0 + half2) * Ncols + coln];
    b.y = B[(k0 + half2 + 1) * Ncols + coln];
    acc = __builtin_amdgcn_wmma_f32_16x16x4_f32(false, a, false, b, (short)0, acc,
                                                false, false);
  }
  float bv = bias[coln];
  int mbase = tm * 16 + ((lane >> 4) << 3);  // VGPR i holds M = i + 8*(lane>=16)
  for (int i = 0; i < 8; ++i) {
    float x = acc[i] + bv;
    if (do_gelu) x = 0.5f * x * (1.0f + erff(x * 0.70710678118654752f));
    D[(mbase + i) * Ncols + coln] = x;
  }
}

// ---------------- host-side orchestration ----------------
extern "C" void kernel_launch(void* const* d_in, const int* in_sizes, int n_in,
                              void* d_out, int out_size, void* d_ws, size_t ws_size,
                              hipStream_t stream) {
  const int*   ei     = (const int*)d_in[0];     // edge_index (2 x NE), int32
  const float* Wsign  = (const float*)d_in[2];   // 8x8
  const float* bsign  = (const float*)d_in[3];   // 8
  const float* Wenc1  = (const float*)d_in[4];   // 8x64
  const float* benc1  = (const float*)d_in[5];   // 64
  const float* Wenc2  = (const float*)d_in[6];   // 64x64
  const float* benc2  = (const float*)d_in[7];   // 64
  float*       out    = (float*)d_out;           // 4096x64

  float* w = (float*)d_ws;
  float* deg  = w;                      // 4096
  float* dis  = deg  + NN;              // 4096
  float* Q    = dis  + NN;              // NN*BS
  float* Z    = Q    + NN * BS;         // NN*BS
  float* G    = Z    + NN * BS;         // 256
  float* Uinv = G    + BS * BS;         // 256
  float* T    = Uinv + BS * BS;         // 256
  float* Ysel = T    + BS * BS;         // 128
  float* pe2  = Ysel + BS * KPE;        // NN*KPE
  float* h    = pe2  + NN * KPE;        // NN*HID

  // --- graph Laplacian pieces ---
  lappe_zerof<<<(NN + 255) / 256, 256, 0, stream>>>(deg, NN);
  lappe_deg<<<NE / 256, 256, 0, stream>>>(ei, deg);
  lappe_dis<<<(NN + 255) / 256, 256, 0, stream>>>(deg, dis);
  lappe_initq<<<NN * BS / 256, 256, 0, stream>>>(Q);

  // --- block subspace iteration on M = 2I - L (sparse) ---
  for (int it = 0; it < ITERS; ++it) {
    lappe_copyf<<<NN * BS / 256, 256, 0, stream>>>(Q, Z, NN * BS);
    lappe_spmv<<<NE * BS / 256, 256, 0, stream>>>(ei, dis, Q, Z);
    lappe_zerof<<<1, 256, 0, stream>>>(G, BS * BS);
    lappe_gram2<<<NN / 128, 256, 0, stream>>>(Z, Z, G);
    lappe_cholinv<<<1, 1, 0, stream>>>(G, Uinv);
    lappe_apply<<<NN / 256, 256, 0, stream>>>(Z, Uinv, Q);
  }

  // --- Rayleigh-Ritz extraction of eigvecs 1..8 ---
  lappe_copyf<<<NN * BS / 256, 256, 0, stream>>>(Q, Z, NN * BS);
  lappe_spmv<<<NE * BS / 256, 256, 0, stream>>>(ei, dis, Q, Z);
  lappe_zerof<<<1, 256, 0, stream>>>(T, BS * BS);
  lappe_gram2<<<NN / 128, 256, 0, stream>>>(Q, Z, T);
  lappe_rrsolve<<<1, 1, 0, stream>>>(T, Ysel);

  // --- SignNet + encoder MLP (WMMA fp32) ---
  lappe_signnet<<<NN / 256, 256, 0, stream>>>(Q, Ysel, Wsign, bsign, pe2);
  // h = gelu(pe2 @ Wenc1 + benc1): tiles = (4096/16)*(64/16) = 1024, 8 waves/block
  lappe_wmma_gemm<<<(NN / 16) * (HID / 16) / 8, 256, 0, stream>>>(pe2, Wenc1, benc1, h,
                                                                  KPE, HID, 1);
  // out = h @ Wenc2 + benc2
  lappe_wmma_gemm<<<(NN / 16) * (HID / 16) / 8, 256, 0, stream>>>(h, Wenc2, benc2, out,
                                                                  HID, HID, 0);
}